// StockPriceLSTM_43868795962061
// MI455X (gfx1250) — compile-verified
//
#include <hip/hip_runtime.h>
#include <math.h>

// Problem constants (from reference)
#define B_   1024
#define T_   512
#define I_   16
#define H_   50
#define G_   200   // 4*H gates
#define GP   208   // gates padded to 13*16
#define NT   13    // N tiles of 16
#define K0   68    // layer0 K = H(50)+I(16) padded to multiple of 4
#define KS0  17    // K0/4 wmma k-steps
#define K1   100   // layer1 K = H+H (already multiple of 4)
#define KS1  25
#define A0S  72    // activation row stride (floats), layer0
#define A1S  104   // activation row stride, layer1
#define GS   212   // gate buffer row stride (conflict-avoiding pad)

#define NW0  (KS0*NT*64)   // 14144 floats of swizzled layer0 weights
#define NW1  (KS1*NT*64)   // 20800 floats of swizzled layer1 weights

typedef float v2f __attribute__((ext_vector_type(2)));
typedef float v8f __attribute__((ext_vector_type(8)));

__device__ __forceinline__ float sigm(float v) {
    return 1.0f / (1.0f + __expf(-v));
}

// ---------------------------------------------------------------------------
// Prep: build combined [W_hh ; W_ih] weight matrices pre-swizzled into the
// per-lane VGPR layout of the WMMA B operand, plus fused biases, in workspace.
//   B-frag layout (16x16x4 f32): lane n<16 holds {B[k0][n], B[k0+1][n]},
//   lane n+16 holds {B[k0+2][n], B[k0+3][n]}.
// ---------------------------------------------------------------------------
__global__ __launch_bounds__(256) void lstm_prep(
    const float* __restrict__ w_ih0, const float* __restrict__ w_hh0,
    const float* __restrict__ b_ih0, const float* __restrict__ b_hh0,
    const float* __restrict__ w_ih1, const float* __restrict__ w_hh1,
    const float* __restrict__ b_ih1, const float* __restrict__ b_hh1,
    float* __restrict__ ws)
{
    int idx = blockIdx.x * 256 + threadIdx.x;
    if (idx < NW0) {
        int frag = idx >> 6, pos = idx & 63;
        int lane = pos >> 1, e = pos & 1;
        int kk = frag / NT, nn = frag % NT;
        int k = kk * 4 + 2 * (lane >> 4) + e;
        int g = nn * 16 + (lane & 15);
        float v = 0.0f;
        if (g < G_) {
            if (k < H_)            v = w_hh0[g * H_ + k];
            else if (k < H_ + I_)  v = w_ih0[g * I_ + (k - H_)];
        }
        ws[idx] = v;
    } else if (idx < NW0 + NW1) {
        int r = idx - NW0;
        int frag = r >> 6, pos = r & 63;
        int lane = pos >> 1, e = pos & 1;
        int kk = frag / NT, nn = frag % NT;
        int k = kk * 4 + 2 * (lane >> 4) + e;
        int g = nn * 16 + (lane & 15);
        float v = 0.0f;
        if (g < G_) {
            if (k < H_) v = w_hh1[g * H_ + k];
            else        v = w_ih1[g * H_ + (k - H_)];
        }
        ws[idx] = v;
    } else if (idx < NW0 + NW1 + GP) {
        int g = idx - NW0 - NW1;
        ws[idx] = (g < G_) ? (b_ih0[g] + b_hh0[g]) : 0.0f;
    } else if (idx < NW0 + NW1 + 2 * GP) {
        int g = idx - NW0 - NW1 - GP;
        ws[idx] = (g < G_) ? (b_ih1[g] + b_hh1[g]) : 0.0f;
    }
}

// ---------------------------------------------------------------------------
// Fused 2-layer LSTM scan. One workgroup = 16 batch rows, 4 waves share the
// 13 gate-column tiles. Weights live in LDS (CDNA5: 320 KB/WGP), reused for
// all 512 steps. All matmuls are V_WMMA_F32_16X16X4_F32 (full fp32).
// A fragments are hoisted (invariant across n-tiles); B fragments are
// batch-loaded into a register array so the DS loads overlap instead of
// serializing one s_wait_dscnt per WMMA.
// ---------------------------------------------------------------------------
__global__ __launch_bounds__(128) void lstm_scan(
    const float* __restrict__ x,     // [B,T,I]
    const float* __restrict__ ws,    // swizzled weights + biases
    float* __restrict__ last)        // [B,H] final layer-1 hidden
{
    extern __shared__ float lds[];
    float* wl0 = lds;                  // NW0
    float* wl1 = wl0 + NW0;            // NW1
    float* bl0 = wl1 + NW1;            // GP
    float* bl1 = bl0 + GP;             // GP
    float* A0  = bl1 + GP;             // 16 x A0S : [h0 (50) | x_t (16) | pad]
    float* A1  = A0 + 16 * A0S;        // 16 x A1S : [h1 (50) | h0_t (50) | pad]
    float* gb  = A1 + 16 * A1S;        // 16 x GS gate buffer
    float* c0  = gb + 16 * GS;         // 16 x H cell state, layer0
    float* c1  = c0 + 16 * H_;         // 16 x H cell state, layer1

    const int tid   = threadIdx.x;
    const int lane  = tid & 31;
    const int wave  = tid >> 5;
    const int lhalf = lane >> 4;   // selects K sub-pair of the fragment
    const int llo   = lane & 15;   // M row (A) / N col (B,C)
    const int b0    = blockIdx.x * 16;

    // Stage weights + biases into LDS (float4 vectorized), zero all state.
    {
        const float4* src = (const float4*)ws;
        float4* dst = (float4*)lds;
        const int n4 = (NW0 + NW1 + 2 * GP) >> 2;
        for (int i = tid; i < n4; i += 128) dst[i] = src[i];
    }
    for (int i = tid; i < 16 * A0S; i += 128) A0[i] = 0.0f;
    for (int i = tid; i < 16 * A1S; i += 128) A1[i] = 0.0f;
    for (int i = tid; i < 16 * H_;  i += 128) { c0[i] = 0.0f; c1[i] = 0.0f; }
    __syncthreads();

    // Load x_0 into A0 columns [H_, H_+I_)
    {
        int row = tid >> 3, pair = tid & 7;
        float2 xv = *(const float2*)(x + ((size_t)(b0 + row) * T_) * I_ + pair * 2);
        *(float2*)(A0 + row * A0S + H_ + pair * 2) = xv;
    }
    __syncthreads();

    const float* arow0 = A0 + llo * A0S + 2 * lhalf;
    const float* arow1 = A1 + llo * A1S + 2 * lhalf;

    for (int t = 0; t < T_; ++t) {
        // ---- GEMM layer0: gates = [h0|x_t] * [W_hh0;W_ih0] + bias ----
        {
            v2f a[KS0];
            #pragma unroll
            for (int kk = 0; kk < KS0; ++kk)
                a[kk] = *(const v2f*)(arow0 + kk * 4);
            for (int nn = wave; nn < NT; nn += 4) {
                v2f bf[KS0];
                #pragma unroll
                for (int kk = 0; kk < KS0; ++kk)
                    bf[kk] = *(const v2f*)(wl0 + (kk * NT + nn) * 64 + lane * 2);
                float bv = bl0[nn * 16 + llo];
                v8f acc = {bv, bv, bv, bv, bv, bv, bv, bv};
                #pragma unroll
                for (int kk = 0; kk < KS0; ++kk)
                    acc = __builtin_amdgcn_wmma_f32_16x16x4_f32(
                              false, a[kk], false, bf[kk], (short)0, acc, false, false);
                float* gout = gb + (8 * lhalf) * GS + nn * 16 + llo;
                #pragma unroll
                for (int j = 0; j < 8; ++j)
                    gout[j * GS] = acc[j];
            }
        }
        __syncthreads();

        // ---- elementwise layer0: i,f,g,o -> c0,h0 ----
        for (int i = tid; i < 16 * H_; i += 128) {
            int m = i / H_, h = i - m * H_;
            const float* gr = gb + m * GS + h;
            float ig = sigm(gr[0]);
            float fg = sigm(gr[H_]);
            float gg = tanhf(gr[2 * H_]);
            float og = sigm(gr[3 * H_]);
            float c  = fg * c0[i] + ig * gg;
            c0[i] = c;
            float hn = og * tanhf(c);
            A0[m * A0S + h]      = hn;   // layer0 recurrent input
            A1[m * A1S + H_ + h] = hn;   // layer1 feed-forward input
        }
        __syncthreads();

        // ---- GEMM layer1: gates = [h1|h0_t] * [W_hh1;W_ih1] + bias ----
        {
            v2f a[KS1];
            #pragma unroll
            for (int kk = 0; kk < KS1; ++kk)
                a[kk] = *(const v2f*)(arow1 + kk * 4);
            for (int nn = wave; nn < NT; nn += 4) {
                v2f bf[KS1];
                #pragma unroll
                for (int kk = 0; kk < KS1; ++kk)
                    bf[kk] = *(const v2f*)(wl1 + (kk * NT + nn) * 64 + lane * 2);
                float bv = bl1[nn * 16 + llo];
                v8f acc = {bv, bv, bv, bv, bv, bv, bv, bv};
                #pragma unroll
                for (int kk = 0; kk < KS1; ++kk)
                    acc = __builtin_amdgcn_wmma_f32_16x16x4_f32(
                              false, a[kk], false, bf[kk], (short)0, acc, false, false);
                float* gout = gb + (8 * lhalf) * GS + nn * 16 + llo;
                #pragma unroll
                for (int j = 0; j < 8; ++j)
                    gout[j * GS] = acc[j];
            }
        }
        __syncthreads();

        // ---- elementwise layer1 + prefetch x_{t+1} ----
        for (int i = tid; i < 16 * H_; i += 128) {
            int m = i / H_, h = i - m * H_;
            const float* gr = gb + m * GS + h;
            float ig = sigm(gr[0]);
            float fg = sigm(gr[H_]);
            float gg = tanhf(gr[2 * H_]);
            float og = sigm(gr[3 * H_]);
            float c  = fg * c1[i] + ig * gg;
            c1[i] = c;
            A1[m * A1S + h] = og * tanhf(c);
        }
        if (t + 1 < T_) {
            int row = tid >> 3, pair = tid & 7;
            float2 xv = *(const float2*)(x + ((size_t)(b0 + row) * T_ + (t + 1)) * I_ + pair * 2);
            *(float2*)(A0 + row * A0S + H_ + pair * 2) = xv;
        }
        __syncthreads();
    }

    // Emit final layer-1 hidden state
    for (int i = tid; i < 16 * H_; i += 128) {
        int m = i / H_, h = i - m * H_;
        last[(size_t)(b0 + m) * H_ + h] = A1[m * A1S + h];
    }
}

// ---------------------------------------------------------------------------
// MLP head: out[b] = fc2_w . relu(fc1_w . last[b] + fc1_b) + fc2_b
// One block of 64 threads per batch element (thread j owns hidden unit j).
// ---------------------------------------------------------------------------
__global__ __launch_bounds__(64) void lstm_fc(
    const float* __restrict__ last,
    const float* __restrict__ fc1_w, const float* __restrict__ fc1_b,
    const float* __restrict__ fc2_w, const float* __restrict__ fc2_b,
    float* __restrict__ out)
{
    __shared__ float red[64];
    const int b = blockIdx.x, j = threadIdx.x;
    const float* lb = last + (size_t)b * H_;
    float s = fc1_b[j];
    #pragma unroll
    for (int h = 0; h < H_; ++h) s += fc1_w[j * H_ + h] * lb[h];
    red[j] = fc2_w[j] * fmaxf(s, 0.0f);
    __syncthreads();
    if (j == 0) {
        float acc = fc2_b[0];
        #pragma unroll
        for (int k = 0; k < 64; ++k) acc += red[k];
        out[b] = acc;
    }
}

extern "C" void kernel_launch(void* const* d_in, const int* in_sizes, int n_in,
                              void* d_out, int out_size, void* d_ws, size_t ws_size,
                              hipStream_t stream)
{
    const float* x     = (const float*)d_in[0];
    const float* w_ih0 = (const float*)d_in[1];
    const float* w_hh0 = (const float*)d_in[2];
    const float* b_ih0 = (const float*)d_in[3];
    const float* b_hh0 = (const float*)d_in[4];
    const float* w_ih1 = (const float*)d_in[5];
    const float* w_hh1 = (const float*)d_in[6];
    const float* b_ih1 = (const float*)d_in[7];
    const float* b_hh1 = (const float*)d_in[8];
    const float* fc1_w = (const float*)d_in[9];
    const float* fc1_b = (const float*)d_in[10];
    const float* fc2_w = (const float*)d_in[11];
    const float* fc2_b = (const float*)d_in[12];
    float* ws  = (float*)d_ws;
    float* out = (float*)d_out;

    float* lastbuf = ws + NW0 + NW1 + 2 * GP;   // [B, H]

    const int prepN = NW0 + NW1 + 2 * GP;
    lstm_prep<<<(prepN + 255) / 256, 256, 0, stream>>>(
        w_ih0, w_hh0, b_ih0, b_hh0, w_ih1, w_hh1, b_ih1, b_hh1, ws);

    const size_t ldsBytes =
        (size_t)(NW0 + NW1 + 2 * GP + 16 * A0S + 16 * A1S + 16 * GS + 2 * 16 * H_)
        * sizeof(float);   // ~168.6 KB, fits CDNA5's 320 KB WGP LDS
    (void)hipFuncSetAttribute((const void*)lstm_scan,
                              hipFuncAttributeMaxDynamicSharedMemorySize,
                              (int)ldsBytes);
    lstm_scan<<<B_ / 16, 128, ldsBytes, stream>>>(x, ws, lastbuf);

    lstm_fc<<<B_, 64, 0, stream>>>(lastbuf, fc1_w, fc1_b, fc2_w, fc2_b, out);
}